// OnlineTripletLoss_30855045055107
// MI455X (gfx1250) — compile-verified
//
#include <hip/hip_runtime.h>
#include <hip/hip_bf16.h>
#include <math.h>

// Problem constants (match reference)
#define BB 512
#define DD 128
#define MARGIN 0.3f

typedef __attribute__((ext_vector_type(2))) float v2f;
typedef __attribute__((ext_vector_type(8))) float v8f;

// ---------------------------------------------------------------------------
// Kernel 0: sq[i] = sum_k emb[i][k]^2
// ---------------------------------------------------------------------------
__global__ void sq_kernel(const float* __restrict__ emb, float* __restrict__ sq) {
    int i = blockIdx.x * blockDim.x + threadIdx.x;
    if (i < BB) {
        const float* row = emb + i * DD;
        float s = 0.0f;
        #pragma unroll 4
        for (int k = 0; k < DD; ++k) { float v = row[k]; s += v * v; }
        sq[i] = s;
    }
}

// ---------------------------------------------------------------------------
// Kernel 1: dist[i][j] = sqrt(max(sq[i]+sq[j]-2*(E E^T)[i][j], 1e-12))
// One wave (32 lanes) computes one 16x16 tile via V_WMMA_F32_16X16X4_F32,
// accumulating K=128 in 32 steps. B = E^T so both A and B tiles are row-loads
// of E (no transpose needed).
//
// fp32 A (16x4) lane layout (ISA 7.12.2): lanes 0-15 hold M=lane, K={0,1} in
// v[0],v[1]; lanes 16-31 hold M=lane-16, K={2,3}. B (4x16) mirrors with N.
// D layout: VGPR r -> row = r + 8*(lane>=16), col = lane&15.
// ---------------------------------------------------------------------------
__global__ void dist_wmma_kernel(const float* __restrict__ emb,
                                 const float* __restrict__ sq,
                                 float* __restrict__ dist) {
    const int lane = threadIdx.x;        // 0..31 (wave32)
    const int half = lane >> 4;          // 0 or 1 -> K offset pair
    const int idx  = lane & 15;          // M (for A) / N (for B)
    const int rowA = blockIdx.x * 16 + idx;
    const int rowB = blockIdx.y * 16 + idx;
    const int koff = half * 2;

    const float* __restrict__ pa = emb + rowA * DD + koff;
    const float* __restrict__ pb = emb + rowB * DD + koff;

    v8f c = {0.f, 0.f, 0.f, 0.f, 0.f, 0.f, 0.f, 0.f};
    #pragma unroll
    for (int k = 0; k < DD; k += 4) {
        v2f a = *(const v2f*)(pa + k);   // K = k+koff, k+koff+1
        v2f b = *(const v2f*)(pb + k);
        // 8 args: (neg_a, A, neg_b, B, c_mod, C, reuse_a, reuse_b)
        c = __builtin_amdgcn_wmma_f32_16x16x4_f32(
                false, a, false, b, (short)0, c, false, false);
    }

    const int col  = blockIdx.y * 16 + idx;
    const float sqc = sq[col];
    #pragma unroll
    for (int r = 0; r < 8; ++r) {
        int row = blockIdx.x * 16 + r + half * 8;
        float d2 = sq[row] + sqc - 2.0f * c[r];
        dist[row * BB + col] = sqrtf(fmaxf(d2, 1e-12f));
    }
}

// ---------------------------------------------------------------------------
// Kernel 2: semi-hard mining. One block per anchor i; dist row + labels in
// LDS. Fixed-order tree reductions -> deterministic partial sum/count.
// ---------------------------------------------------------------------------
__global__ void mine_kernel(const float* __restrict__ dist,
                            const int* __restrict__ labels,
                            float* __restrict__ psum,
                            float* __restrict__ pcnt) {
    __shared__ float drow[BB];
    __shared__ int   lab[BB];
    __shared__ float red_a[256];
    __shared__ float red_b[256];

    const int i   = blockIdx.x;
    const int tid = threadIdx.x;         // 256 threads

    for (int j = tid; j < BB; j += 256) {
        drow[j] = dist[i * BB + j];
        lab[j]  = labels[j];
    }
    __syncthreads();

    const int li = lab[i];

    // min over negatives (fill POS_BIG = 1e30 like reference)
    float lmin = 1e30f;
    for (int k = tid; k < BB; k += 256)
        if (lab[k] != li) lmin = fminf(lmin, drow[k]);
    red_a[tid] = lmin;
    __syncthreads();
    for (int s = 128; s > 0; s >>= 1) {
        if (tid < s) red_a[tid] = fminf(red_a[tid], red_a[tid + s]);
        __syncthreads();
    }
    const float min_neg = red_a[0];
    const bool  has_neg = (min_neg < 1e29f);
    __syncthreads();

    float sum = 0.0f, cnt = 0.0f;
    if (has_neg) {
        for (int j = tid; j < BB; j += 256) {
            if (j == i || lab[j] != li) continue;   // positives only
            const float d_ap = drow[j];
            const float hi   = d_ap + MARGIN;
            float semi_max = -1e30f;                // NEG_BIG
            for (int k = 0; k < BB; ++k) {
                const float d_an = drow[k];
                const bool semi = (lab[k] != li) && (d_an > d_ap) && (d_an < hi);
                semi_max = semi ? fmaxf(semi_max, d_an) : semi_max;
            }
            const float d_neg = (semi_max > -1e29f) ? semi_max : min_neg;
            const float pp = d_ap - d_neg + MARGIN;
            sum += (pp > 0.0f) ? pp : 0.0f;
            cnt += 1.0f;
        }
    }

    red_a[tid] = sum;
    red_b[tid] = cnt;
    __syncthreads();
    for (int s = 128; s > 0; s >>= 1) {
        if (tid < s) { red_a[tid] += red_a[tid + s]; red_b[tid] += red_b[tid + s]; }
        __syncthreads();
    }
    if (tid == 0) { psum[i] = red_a[0]; pcnt[i] = red_b[0]; }
}

// ---------------------------------------------------------------------------
// Kernel 3: final deterministic reduction -> scalar loss
// ---------------------------------------------------------------------------
__global__ void finalize_kernel(const float* __restrict__ psum,
                                const float* __restrict__ pcnt,
                                float* __restrict__ out) {
    __shared__ float ss[BB];
    __shared__ float cc[BB];
    const int tid = threadIdx.x;         // 512 threads
    ss[tid] = psum[tid];
    cc[tid] = pcnt[tid];
    __syncthreads();
    for (int s = 256; s > 0; s >>= 1) {
        if (tid < s) { ss[tid] += ss[tid + s]; cc[tid] += cc[tid + s]; }
        __syncthreads();
    }
    if (tid == 0) {
        const float c = cc[0];
        out[0] = (c > 0.0f) ? (ss[0] / c) : 0.0f;
    }
}

// ---------------------------------------------------------------------------
extern "C" void kernel_launch(void* const* d_in, const int* in_sizes, int n_in,
                              void* d_out, int out_size, void* d_ws, size_t ws_size,
                              hipStream_t stream) {
    const float* emb    = (const float*)d_in[0];   // (512,128) fp32
    const int*   labels = (const int*)d_in[1];     // (512,) int
    float* out = (float*)d_out;

    float* dist = (float*)d_ws;                    // 512*512 fp32 = 1 MB
    float* sq   = dist + BB * BB;                  // 512
    float* psum = sq + BB;                         // 512
    float* pcnt = psum + BB;                       // 512

    sq_kernel<<<(BB + 127) / 128, 128, 0, stream>>>(emb, sq);

    dim3 grid(BB / 16, BB / 16);                   // 32x32 tiles, 1 wave each
    dist_wmma_kernel<<<grid, 32, 0, stream>>>(emb, sq, dist);

    mine_kernel<<<BB, 256, 0, stream>>>(dist, labels, psum, pcnt);

    finalize_kernel<<<1, BB, 0, stream>>>(psum, pcnt, out);
}